// VGP_47991964565576
// MI455X (gfx1250) — compile-verified
//
#include <hip/hip_runtime.h>
#include <hip/hip_bf16.h>
#include <math.h>

typedef __attribute__((ext_vector_type(2))) float v2f;
typedef __attribute__((ext_vector_type(4))) float v4f;
typedef __attribute__((ext_vector_type(8))) float v8f;

#define BATCH   5
#define KSPLIT  16
#define TNB     128     // output columns per block (2 x 16 per wave, 4 waves)
#define SROW    136     // padded LDS row stride (floats): halves hit disjoint banks
#define TKC     32      // K-chunk staged in LDS
#define GT      128     // 4 waves per block
#define PB      64      // partial blocks for loss reductions

#define USE_ASYNC_LDS 1 // CDNA5 global_load_async_to_lds_* staging (no VGPR round trip)

// ---------------------------------------------------------------------------
// Skinny GEMM: Cpart[ks][m][n] = sum_{k in split ks} A[m][k] * W[k][n]
// A: [5,K] fp32 row-major. Rows 5..15 of the 16x16 WMMA tile are fed with a
// CLAMPED row pointer (row 4 repeated): garbage only lands in output rows
// 5..15, which are never stored.  Same trick for columns >= N.
// Uses V_WMMA_F32_16X16X4_F32 (f32 layouts per ISA 7.12.2):
//   A 16x4 : v[0] lanes0-15=(M=lane,K=0), lanes16-31=(M,K=2); v[1]: K=1/K=3
//   B 4x16 : v[0] lanes0-15=(K=0,N=lane), lanes16-31=(K=2,N); v[1]: K=1/K=3
//   C 16x16: v[r]  lanes0-15=(M=r,N=lane), lanes16-31=(M=r+8,N)
// Weight staging: GLOBAL_LOAD_ASYNC_TO_LDS_B128 (ASYNCcnt), producer waits
// s_wait_asynccnt 0 before the barrier; consumers read after the barrier.
// ---------------------------------------------------------------------------
__global__ void gemm_skinny_wmma(const float* __restrict__ A,
                                 const float* __restrict__ W,
                                 float* __restrict__ Cp,
                                 int K, int N, int klen)
{
    __shared__ float sB[TKC * SROW];
    const int tid  = threadIdx.x;
    const int lane = tid & 31;
    const int wid  = tid >> 5;
    const int n0   = blockIdx.x * TNB;
    const int k0   = blockIdx.y * klen;

    const int m    = lane & 15;
    const int koff = (lane >> 4) << 1;                 // 0 | 2
    const int rowm = (m < BATCH) ? m : (BATCH - 1);    // clamped row (branchless)

    const size_t sN = (size_t)N;

    // A fragment pointer: lane reads A[rowm][k0+koff + kk], bumped by TKC/chunk
    const float* ap = A + (size_t)rowm * K + k0 + koff;

    // B fragment LDS base: row koff, col = wid*32 + m (tile0), +16 (tile1)
    const float* bp = &sB[koff * SROW + (wid << 5) + m];

    // --- staging pointers (hoisted; bumped by constant strides) -----------
    // vectorized path: thread owns col 4*(tid&31), rows (tid>>5)+4p, p=0..7
    const float* wv = W + (size_t)(k0 + (tid >> 5)) * sN + n0 + ((tid & 31) << 2);
    float*       sv = &sB[(tid >> 5) * SROW + ((tid & 31) << 2)];
    // scalar path (N=4097): one clamped column per thread, rows 0..31
    int ncl = n0 + tid; if (ncl > N - 1) ncl = N - 1;
    const float* wsc = W + (size_t)k0 * sN + ncl;
    float*       ss  = &sB[tid];

    // LDS byte offsets for async staging (generic LDS addr low 32 bits)
    const unsigned lv = (unsigned)(uintptr_t)sv;
    const unsigned ls = (unsigned)(uintptr_t)ss;

    v8f acc0 = {0.f,0.f,0.f,0.f,0.f,0.f,0.f,0.f};
    v8f acc1 = {0.f,0.f,0.f,0.f,0.f,0.f,0.f,0.f};

    const bool vec = ((N & 3) == 0);

    for (int kc = 0; kc < klen; kc += TKC) {
        if (vec) {
            const float* wp = wv;
#if USE_ASYNC_LDS
            unsigned     lo = lv;
            #pragma unroll
            for (int p = 0; p < 8; ++p) {              // 32 rows x 128 cols
                asm volatile("global_load_async_to_lds_b128 %0, %1, off"
                             :: "v"(lo), "v"(wp) : "memory");
                wp += 4 * sN;
                lo += 4 * SROW * 4;
            }
#else
            float*       sp = sv;
            #pragma unroll
            for (int p = 0; p < 8; ++p) {
                v4f v = __builtin_nontemporal_load((const v4f*)wp);
                *(v4f*)sp = v;
                wp += 4 * sN;
                sp += 4 * SROW;
            }
#endif
            wv += (size_t)TKC * sN;
        } else {
            const float* wp = wsc;
#if USE_ASYNC_LDS
            unsigned     lo = ls;
            #pragma unroll
            for (int r = 0; r < TKC; ++r) {
                asm volatile("global_load_async_to_lds_b32 %0, %1, off"
                             :: "v"(lo), "v"(wp) : "memory");
                wp += sN;
                lo += SROW * 4;
            }
#else
            float*       sp = ss;
            #pragma unroll
            for (int r = 0; r < TKC; ++r) {
                *sp = __builtin_nontemporal_load(wp);
                wp += sN;
                sp += SROW;
            }
#endif
            wsc += (size_t)TKC * sN;
        }
#if USE_ASYNC_LDS
        asm volatile("s_wait_asynccnt 0x0" ::: "memory");
#endif
        __syncthreads();

        #pragma unroll
        for (int kk = 0; kk < TKC; kk += 4) {
            v2f av = *(const v2f*)(ap + kk);           // one b64: K=koff,koff+1
            const float* bb = bp + kk * SROW;
            v2f b0, b1;
            b0[0] = bb[0];    b0[1] = bb[SROW];        // tile 0
            b1[0] = bb[16];   b1[1] = bb[SROW + 16];   // tile 1 (same A frag)
            acc0 = __builtin_amdgcn_wmma_f32_16x16x4_f32(
                       false, av, false, b0, (short)0, acc0, false, false);
            acc1 = __builtin_amdgcn_wmma_f32_16x16x4_f32(
                       false, av, false, b1, (short)0, acc1, false, false);
        }
        ap += TKC;
        __syncthreads();
    }

    // store rows M=0..4 (lanes 0-15 hold M=r, N=lane)
    if (lane < 16) {
        float* outp = Cp + (size_t)blockIdx.y * BATCH * N;
        const int na = n0 + (wid << 5) + lane;
        const int nb = na + 16;
        if (na < N) {
            outp[0*sN + na] = acc0[0];
            outp[1*sN + na] = acc0[1];
            outp[2*sN + na] = acc0[2];
            outp[3*sN + na] = acc0[3];
            outp[4*sN + na] = acc0[4];
        }
        if (nb < N) {
            outp[0*sN + nb] = acc1[0];
            outp[1*sN + nb] = acc1[1];
            outp[2*sN + nb] = acc1[2];
            outp[3*sN + nb] = acc1[3];
            outp[4*sN + nb] = acc1[4];
        }
    }
}

// fold split-K partials + bias (+ optional ReLU)
__global__ void reduce_bias(const float* __restrict__ Cp,
                            const float* __restrict__ bias,
                            float* __restrict__ out, int N, int act)
{
    int idx = blockIdx.x * blockDim.x + threadIdx.x;
    int total = BATCH * N;
    if (idx >= total) return;
    int n = idx % N;
    float s = bias[n];
    #pragma unroll
    for (int k = 0; k < KSPLIT; ++k) s += Cp[(size_t)k * total + idx];
    if (act) s = fmaxf(s, 0.0f);
    out[idx] = s;
}

// ---------------------------------------------------------------------------
// Elementwise / GP / loss kernels
// ---------------------------------------------------------------------------
__global__ void reparam_nm(const float* __restrict__ mean,
                           const float* __restrict__ logcov,
                           const float* __restrict__ eps,
                           float* __restrict__ out, int n)
{
    int i = blockIdx.x * blockDim.x + threadIdx.x;
    if (i < n) out[i] = expf(0.5f * logcov[i]) * eps[i] + mean[i];
}

__global__ void gp_pairs(const float* __restrict__ s,
                         const float* __restrict__ xi,
                         const float* __restrict__ w,
                         const float* __restrict__ sigma,
                         float* __restrict__ Kss, float* __restrict__ Kxs)
{
    __shared__ float red[256];
    const int b = blockIdx.x;                 // 0..24 -> Kss, 25..49 -> Kxs
    const int p = (b < 25) ? b : b - 25;
    const int i = p / 5, j = p % 5;
    const float* ai = ((b < 25) ? s : xi) + (size_t)i * 4096;
    const float* bj = s + (size_t)j * 4096;
    float acc = 0.f;
    for (int t = threadIdx.x; t < 4096; t += 256) {
        float d = ai[t] - bj[t];
        acc += w[t] * d * d;
    }
    red[threadIdx.x] = acc; __syncthreads();
    for (int o = 128; o > 0; o >>= 1) {
        if (threadIdx.x < o) red[threadIdx.x] += red[threadIdx.x + o];
        __syncthreads();
    }
    if (threadIdx.x == 0) {
        float sg = sigma[0];
        float Kv = sg * sg * expf(-0.5f * red[0]);
        if (b < 25) Kss[p] = Kv; else Kxs[p] = Kv;
    }
}

__global__ void gp_solve(const float* __restrict__ Kss,
                         const float* __restrict__ Kxs,
                         const float* __restrict__ t,
                         const float* __restrict__ eps_f,
                         const float* __restrict__ sigma,
                         float* __restrict__ f,
                         float* __restrict__ qf_mean,
                         float* __restrict__ logqf)
{
    if (threadIdx.x != 0 || blockIdx.x != 0) return;
    float M[5][10];
    for (int i = 0; i < 5; ++i)
        for (int j = 0; j < 5; ++j) {
            M[i][j] = Kss[i * 5 + j];
            M[i][5 + j] = (i == j) ? 1.0f : 0.0f;
        }
    for (int c = 0; c < 5; ++c) {
        int piv = c; float best = fabsf(M[c][c]);
        for (int r = c + 1; r < 5; ++r)
            if (fabsf(M[r][c]) > best) { best = fabsf(M[r][c]); piv = r; }
        if (piv != c)
            for (int j = 0; j < 10; ++j) { float tmp = M[c][j]; M[c][j] = M[piv][j]; M[piv][j] = tmp; }
        float inv = 1.0f / M[c][c];
        for (int j = 0; j < 10; ++j) M[c][j] *= inv;
        for (int r = 0; r < 5; ++r) {
            if (r == c) continue;
            float fac = M[r][c];
            for (int j = 0; j < 10; ++j) M[r][j] -= fac * M[c][j];
        }
    }
    float kkinv[5][5];
    for (int i = 0; i < 5; ++i)
        for (int j = 0; j < 5; ++j) {
            float sum = 0.f;
            for (int k = 0; k < 5; ++k) sum += Kxs[i * 5 + k] * M[k][5 + j];
            kkinv[i][j] = sum;
        }
    float sg = sigma[0];
    for (int i = 0; i < 5; ++i) {
        float qm = 0.f, qc = sg * sg;
        for (int j = 0; j < 5; ++j) {
            qm += kkinv[i][j] * t[j];
            qc -= kkinv[i][j] * Kxs[i * 5 + j];
        }
        qf_mean[i] = qm;
        logqf[i]   = logf(qc);
        f[i]       = sqrtf(qc) * eps_f[i] + qm;
    }
}

__global__ void h2_outer(const float* __restrict__ f,
                         const float* __restrict__ W2,
                         const float* __restrict__ b2,
                         float* __restrict__ h2)
{
    int idx = blockIdx.x * blockDim.x + threadIdx.x;
    if (idx >= BATCH * 4096) return;
    int i = idx >> 12, n = idx & 4095;
    h2[idx] = fmaxf(f[i] * W2[n] + b2[n], 0.0f);
}

__global__ void z_concat(const float* __restrict__ zm, const float* __restrict__ zc,
                         const float* __restrict__ eps_z, const float* __restrict__ x,
                         float* __restrict__ z, float* __restrict__ xz)
{
    int idx = blockIdx.x * blockDim.x + threadIdx.x;
    if (idx >= BATCH * 4096) return;
    int i = idx >> 12, n = idx & 4095;
    float zv = expf(0.5f * zc[idx]) * eps_z[idx] + zm[idx];
    z[idx] = zv;
    xz[(size_t)i * 8192 + 4096 + n] = zv;
    xz[(size_t)i * 8192 + n]        = x[idx];
}

__device__ __forceinline__ void block_reduce_store(float acc, float* part)
{
    __shared__ float red[256];
    red[threadIdx.x] = acc; __syncthreads();
    for (int o = 128; o > 0; o >>= 1) {
        if (threadIdx.x < o) red[threadIdx.x] += red[threadIdx.x + o];
        __syncthreads();
    }
    if (threadIdx.x == 0) part[blockIdx.x] = red[0];
}

__global__ void kld1_partial(const float* __restrict__ zm, const float* __restrict__ zc,
                             float* __restrict__ part)
{
    float acc = 0.f;
    for (int i = blockIdx.x * 256 + threadIdx.x; i < BATCH * 4096; i += gridDim.x * 256) {
        float c = zc[i], mn = zm[i];
        acc += -c - 1.0f + expf(c) + mn * mn;
    }
    block_reduce_store(acc, part);
}

__global__ void kld2_partial(const float* __restrict__ qxm, const float* __restrict__ qxc,
                             const float* __restrict__ qfm, const float* __restrict__ lqf,
                             const float* __restrict__ rm,  const float* __restrict__ rc,
                             float* __restrict__ part)
{
    const int NT = BATCH * 4097;
    float acc = 0.f;
    for (int idx = blockIdx.x * 256 + threadIdx.x; idx < NT; idx += gridDim.x * 256) {
        int i = idx / 4097, j = idx - i * 4097;
        float qm = (j < 4096) ? qxm[i * 4096 + j] : qfm[i];
        float qc = (j < 4096) ? qxc[i * 4096 + j] : lqf[i];
        float rmv = rm[idx], rcv = rc[idx];
        float d = qm - rmv;
        acc += rcv - qc - 1.0f + expf(qc - rcv) + d * d * expf(-rcv);
    }
    block_reduce_store(acc, part);
}

__global__ void nll_partial(const float* __restrict__ x, const float* __restrict__ xm,
                            const float* __restrict__ xc, float* __restrict__ part)
{
    const float logpi = 1.14472988584940017f;   // ln(pi)
    float acc = 0.f;
    for (int i = blockIdx.x * 256 + threadIdx.x; i < BATCH * 4096; i += gridDim.x * 256) {
        float c = xc[i];
        float d = x[i] - xm[i];
        acc += logpi + c + expf(-c) * d * d;
    }
    block_reduce_store(acc, part);
}

__global__ void loss_final(const float* __restrict__ p1, const float* __restrict__ p2,
                           const float* __restrict__ p3, float* __restrict__ out)
{
    if (threadIdx.x != 0 || blockIdx.x != 0) return;
    float s1 = 0.f, s2 = 0.f, s3 = 0.f;
    for (int i = 0; i < PB; ++i) { s1 += p1[i]; s2 += p2[i]; s3 += p3[i]; }
    out[0] = 0.5f * (s1 + s2) * 0.2f;   // kld = kld1 + kld2, /B with B=5
    out[1] = 0.5f * s3 * 0.2f;          // nll
}

// ---------------------------------------------------------------------------
extern "C" void kernel_launch(void* const* d_in, const int* in_sizes, int n_in,
                              void* d_out, int out_size, void* d_ws, size_t ws_size,
                              hipStream_t stream)
{
    const float* x      = (const float*)d_in[0];
    const float* eps_xi = (const float*)d_in[1];
    const float* eps_f  = (const float*)d_in[2];
    const float* eps_z  = (const float*)d_in[3];
    const float* W1  = (const float*)d_in[4];  const float* b1  = (const float*)d_in[5];
    const float* W11 = (const float*)d_in[6];  const float* b11 = (const float*)d_in[7];
    const float* W12 = (const float*)d_in[8];  const float* b12 = (const float*)d_in[9];
    const float* W2  = (const float*)d_in[10]; const float* b2  = (const float*)d_in[11];
    const float* W21 = (const float*)d_in[12]; const float* b21 = (const float*)d_in[13];
    const float* W22 = (const float*)d_in[14]; const float* b22 = (const float*)d_in[15];
    const float* W23 = (const float*)d_in[16]; const float* b23 = (const float*)d_in[17];
    const float* W24 = (const float*)d_in[18]; const float* b24 = (const float*)d_in[19];
    const float* W25 = (const float*)d_in[20]; const float* b25 = (const float*)d_in[21];
    const float* W3  = (const float*)d_in[22]; const float* b3  = (const float*)d_in[23];
    const float* W31 = (const float*)d_in[24]; const float* b31 = (const float*)d_in[25];
    const float* W32 = (const float*)d_in[26]; const float* b32 = (const float*)d_in[27];
    const float* s     = (const float*)d_in[28];
    const float* t     = (const float*)d_in[29];
    const float* sigma = (const float*)d_in[30];
    const float* w     = (const float*)d_in[31];

    float* out    = (float*)d_out;
    float* zm_out = out + 2;
    float* zc_out = out + 2 + 20480;
    float* xm_out = out + 2 + 2 * 20480;
    float* xc_out = out + 2 + 3 * 20480;

    float* ws = (float*)d_ws;
    size_t off = 0;
    auto alloc = [&](size_t n) { float* p = ws + off; off += (n + 3) & ~(size_t)3; return p; };

    float* h1   = alloc(20480);
    float* qxm  = alloc(20480);
    float* qxc  = alloc(20480);
    float* xi   = alloc(20480);
    float* h2   = alloc(20480);
    float* z    = alloc(20480);
    float* xz   = alloc(40960);
    float* h23  = alloc(20480);
    float* rm   = alloc(20485);
    float* rc   = alloc(20485);
    float* h3   = alloc(20480);
    float* Kss  = alloc(25);
    float* Kxs  = alloc(25);
    float* fb   = alloc(8);
    float* qfm  = alloc(8);
    float* lqf  = alloc(8);
    float* cp   = alloc((size_t)KSPLIT * BATCH * 4097);   // split-K partials
    float* p1   = alloc(PB);
    float* p2   = alloc(PB);
    float* p3   = alloc(PB);

    auto gemm = [&](const float* A, const float* Wt, const float* bias,
                    float* dst, int K, int N, int act) {
        int klen = K / KSPLIT;                 // 4096 -> 256, 8192 -> 512
        dim3 grid((N + TNB - 1) / TNB, KSPLIT);
        gemm_skinny_wmma<<<grid, GT, 0, stream>>>(A, Wt, cp, K, N, klen);
        int total = BATCH * N;
        reduce_bias<<<(total + 255) / 256, 256, 0, stream>>>(cp, bias, dst, N, act);
    };

    const int EW = (BATCH * 4096 + 255) / 256;

    // encode_1
    gemm(x,  W1,  b1,  h1,  4096, 4096, 1);
    gemm(h1, W11, b11, qxm, 4096, 4096, 0);
    gemm(h1, W12, b12, qxc, 4096, 4096, 0);
    reparam_nm<<<EW, 256, 0, stream>>>(qxm, qxc, eps_xi, xi, BATCH * 4096);

    // reparameterize_gp
    gp_pairs<<<50, 256, 0, stream>>>(s, xi, w, sigma, Kss, Kxs);
    gp_solve<<<1, 32, 0, stream>>>(Kss, Kxs, t, eps_f, sigma, fb, qfm, lqf);

    // encode_2
    h2_outer<<<EW, 256, 0, stream>>>(fb, W2, b2, h2);
    gemm(h2, W21, b21, zm_out, 4096, 4096, 0);
    gemm(h2, W22, b22, zc_out, 4096, 4096, 0);
    z_concat<<<EW, 256, 0, stream>>>(zm_out, zc_out, eps_z, x, z, xz);

    // encode_3
    gemm(xz,  W23, b23, h23, 8192, 4096, 1);
    gemm(h23, W24, b24, rm,  4096, 4097, 0);
    gemm(h23, W25, b25, rc,  4096, 4097, 0);

    // decode
    gemm(z,  W3,  b3,  h3,     4096, 4096, 1);
    gemm(h3, W31, b31, xm_out, 4096, 4096, 0);
    gemm(h3, W32, b32, xc_out, 4096, 4096, 0);

    // losses (deterministic two-stage reductions)
    kld1_partial<<<PB, 256, 0, stream>>>(zm_out, zc_out, p1);
    kld2_partial<<<PB, 256, 0, stream>>>(qxm, qxc, qfm, lqf, rm, rc, p2);
    nll_partial<<<PB, 256, 0, stream>>>(x, xm_out, xc_out, p3);
    loss_final<<<1, 32, 0, stream>>>(p1, p2, p3, out);
}